// WhitePool_28406913696451
// MI455X (gfx1250) — compile-verified
//
#include <hip/hip_runtime.h>

// Problem constants (from reference): x(16,64,256,256) i32, white_table(64,3,256,256) i32,
// last_table(64,256) f32 -> out(16,64,128,128) f32
#define B_    16
#define C_    64
#define H_    256
#define W_    256
#define OH    128
#define OW    128
#define NPIX  (B_ * C_ * OH * OW)   // 16,777,216 output pixels
#define KCH   4                      // independent chains per thread (ILP for gather latency)
#define CHUNK (NPIX / KCH)           // 4,194,304
#define TPB   256                    // 8 wave32s per block

typedef int v2i __attribute__((ext_vector_type(2)));

__global__ __launch_bounds__(TPB) void whitepool_kernel(
    const int*   __restrict__ x,
    const int*   __restrict__ wt,
    const float* __restrict__ lt,
    float*       __restrict__ out)
{
    __shared__ float lt_lds[256];

    const unsigned tid   = threadIdx.x;
    const unsigned bid   = blockIdx.x;
    // 64 blocks cover one (b,c) plane of 128*128 pixels -> plane/channel are
    // functions of blockIdx only (block-uniform, scalar registers).
    const unsigned plane = bid >> 6;        // b*64 + c  (for chunk k=0)
    const unsigned c     = plane & (C_ - 1);

    // ---- Stage last_table[c, 0:256] (1 KB) into LDS via async global->LDS DMA ----
    {
        const float* lt_row = lt + ((size_t)c << 8);            // scalar (SGPR) base
        unsigned lds_off = (unsigned)(size_t)(&lt_lds[tid]);    // low 32 bits == LDS offset
        unsigned voff    = tid << 2;                            // per-lane byte offset
        asm volatile(
            "global_load_async_to_lds_b32 %0, %1, %2\n\t"
            "s_wait_asynccnt 0"
            :
            : "v"(lds_off), "v"(voff), "s"(lt_row)
            : "memory");
    }
    __syncthreads();   // all 8 waves' async stages complete before any lt_lds read

    const unsigned base = bid * TPB + tid;           // pixel index within chunk 0
    const unsigned w    = base & (OW - 1);
    const unsigned h    = (base >> 7) & (OH - 1);

    // x element offset of the 2x2 window's top-left for chunk 0.
    // Chunk k adds k*CHUNK pixels == k*256 planes == k*NPIX x-elements (2^24).
    const size_t xbase = ((size_t)plane << 16) + (size_t)((h << 1) << 8) + (size_t)(w << 1);
    const int* wt_c = wt + (((size_t)c * 3u) << 16);   // channel's 3 tables (3*256KB)

    // ---- Streaming window loads (non-temporal: protect L2 residency of tables) ----
    v2i r0[KCH], r1[KCH];
#pragma unroll
    for (int k = 0; k < KCH; ++k) {
        const int* xp = x + xbase + (size_t)k * (size_t)NPIX;
        r0[k] = __builtin_nontemporal_load((const v2i*)xp);         // x00, x01
        r1[k] = __builtin_nontemporal_load((const v2i*)(xp + W_));  // x10, x11
    }

    // ---- 3-deep dependent gather chain, 4 independent chains interleaved ----
    int o[KCH];
#pragma unroll
    for (int k = 0; k < KCH; ++k) o[k] = r0[k].x;
#pragma unroll
    for (int k = 0; k < KCH; ++k) o[k] = wt_c[(o[k] << 8) + r0[k].y];                  // T0
#pragma unroll
    for (int k = 0; k < KCH; ++k) o[k] = wt_c[(1 << 16) + (o[k] << 8) + r1[k].x];      // T1
#pragma unroll
    for (int k = 0; k < KCH; ++k) o[k] = wt_c[(2 << 16) + (o[k] << 8) + r1[k].y];      // T2

    // ---- Final lookup from LDS, non-temporal streaming store ----
#pragma unroll
    for (int k = 0; k < KCH; ++k) {
        float y = lt_lds[o[k]];
        __builtin_nontemporal_store(y, out + (size_t)base + (size_t)k * (size_t)CHUNK);
    }
}

extern "C" void kernel_launch(void* const* d_in, const int* in_sizes, int n_in,
                              void* d_out, int out_size, void* d_ws, size_t ws_size,
                              hipStream_t stream) {
    const int*   x  = (const int*)d_in[0];    // (16,64,256,256) int32
    const int*   wt = (const int*)d_in[1];    // (64,3,256,256) int32
    const float* lt = (const float*)d_in[2];  // (64,256) float32
    float* out = (float*)d_out;               // (16,64,128,128) float32

    dim3 grid(CHUNK / TPB);                   // 16384 blocks
    whitepool_kernel<<<grid, TPB, 0, stream>>>(x, wt, lt, out);
}